// ResilientSwarmGNN_7876970020998
// MI455X (gfx1250) — compile-verified
//
#include <hip/hip_runtime.h>
#include <hip/hip_bf16.h>
#include <math.h>

// ---------------------------------------------------------------------------
// ResilientSwarmGNN for MI455X (gfx1250, wave32, WMMA)
//
// fp32 WMMA (v_wmma_f32_16x16x4_f32) for all GEMMs. Weights are pre-packed
// into per-lane WMMA B-fragment order so the hot loop issues ONE coalesced
// global_load_b64 per wmma instead of two strided b32 loads.
// Attention branch is analytically constant (mean of softmax == 1/HEADS = .25)
// and is skipped entirely.
// ---------------------------------------------------------------------------

#define N_NODES  50000
#define N_EDGES  800000
#define EMB      64
#define HID      128
#define CATD     136     // 2*EMB + EDGE_DIM
#define EDGE_DIM 8
#define LDA      140     // edge A-tile LDS stride (16B aligned rows, conflict-free frags)
#define LDH      132     // H-tile LDS stride
#define LDN      132     // node A/H tile stride
#define LDY      68      // node Y tile stride
#define LN_EPS   1e-5f

typedef float v2f __attribute__((ext_vector_type(2)));
typedef float v8f __attribute__((ext_vector_type(8)));

__device__ __forceinline__ v8f wmma4(v2f a, v2f b, v8f c) {
  // D = A(16x4) x B(4x16) + C(16x16), fp32
  return __builtin_amdgcn_wmma_f32_16x16x4_f32(false, a, false, b, (short)0, c,
                                               false, false);
}

__device__ __forceinline__ float wave_sum32(float v) {
  #pragma unroll
  for (int off = 16; off > 0; off >>= 1) v += __shfl_xor(v, off, 32);
  return v;
}

// One 16x16 output tile accumulated over kdim: A from LDS, B from packed
// fragment-order weights (one v2f load per wmma, lane-contiguous).
__device__ __forceinline__ v8f wave_gemm_acc(const float* __restrict__ Alds, int lda,
                                             const v2f* __restrict__ Wp, int nsteps,
                                             int nt, int lane) {
  const int half = lane >> 4;
  const int nl   = lane & 15;
  v8f acc = {0.f, 0.f, 0.f, 0.f, 0.f, 0.f, 0.f, 0.f};
  const v2f* wp = Wp + (size_t)nt * nsteps * 32 + lane;
  const float* arow = Alds + nl * lda + 2 * half;
  for (int ks = 0; ks < nsteps; ++ks) {
    v2f a; a.x = arow[ks * 4]; a.y = arow[ks * 4 + 1];
    acc = wmma4(a, wp[ks * 32], acc);
  }
  return acc;
}

// H[16][nout] = relu(A[16][kdim] @ W + bias); W pre-packed; one wave.
__device__ __forceinline__ void wave_gemm_relu(const float* __restrict__ Alds, int lda,
                                               const v2f* __restrict__ Wp,
                                               const float* __restrict__ bias,
                                               int kdim, int nout,
                                               float* __restrict__ Hout, int ldh,
                                               int lane) {
  const int half = lane >> 4;
  const int nl   = lane & 15;
  const int nsteps = kdim >> 2;
  for (int nt = 0; nt < (nout >> 4); ++nt) {
    v8f acc = wave_gemm_acc(Alds, lda, Wp, nsteps, nt, lane);
    const float bs = bias[nt * 16 + nl];
    #pragma unroll
    for (int r = 0; r < 8; ++r) {
      float v = acc[r] + bs;
      Hout[(r + 8 * half) * ldh + (nt * 16 + nl)] = v > 0.f ? v : 0.f;
    }
  }
}

// ---------------------------------------------------------------------------
// Weight packer: W[kdim][nout] row-major -> fragment order
// Wp[((nt*nsteps)+kstep)*64 + lane*2 + t] = W[kstep*4 + 2*(lane>>4) + t][nt*16 + (lane&15)]
// Optional swap of k-rows 0..63 <-> 64..127 (rel_in vs msg_in concat order).
// ---------------------------------------------------------------------------
__global__ void gnn_pack_weights_kernel(const float* __restrict__ W,
                                        float* __restrict__ Wp,
                                        int kdim, int nout, int swap_halves) {
  const int idx = blockIdx.x * blockDim.x + threadIdx.x;
  if (idx >= kdim * nout) return;
  const int nsteps = kdim >> 2;
  const int e = idx & 63;
  const int lane = e >> 1, t = e & 1;
  const int rest = idx >> 6;
  const int kstep = rest % nsteps;
  const int nt    = rest / nsteps;
  const int half = lane >> 4, nl = lane & 15;
  int k = kstep * 4 + 2 * half + t;
  const int n = nt * 16 + nl;
  if (swap_halves) k = (k < 64) ? (k + 64) : (k < 128 ? k - 64 : k);
  Wp[idx] = W[k * nout + n];
}

__global__ void gnn_zero_kernel(float* __restrict__ p, size_t n) {
  const size_t stride = (size_t)gridDim.x * blockDim.x;
  for (size_t i = blockIdx.x * (size_t)blockDim.x + threadIdx.x; i < n; i += stride)
    p[i] = 0.f;
}

// ---------------------------------------------------------------------------
// Encoder: wave-per-node (8 nodes / 256-thread block). VALU; <1% of FLOPs.
// ---------------------------------------------------------------------------
__global__ __launch_bounds__(256) void gnn_encoder_kernel(
    const float* __restrict__ obs, const float* __restrict__ unc,
    const float* __restrict__ obs1w, const float* __restrict__ obs1b,
    const float* __restrict__ obs_ln_g, const float* __restrict__ obs_ln_b,
    const float* __restrict__ obs2w, const float* __restrict__ obs2b,
    const float* __restrict__ unc1w, const float* __restrict__ unc1b,
    const float* __restrict__ unc2w, const float* __restrict__ unc2b,
    const float* __restrict__ fusw, const float* __restrict__ fusb,
    const float* __restrict__ fus_ln_g, const float* __restrict__ fus_ln_b,
    float* __restrict__ xout) {
  __shared__ float obsS[8][25];
  __shared__ float hS[8][128];
  __shared__ float u1S[8][32];
  __shared__ float combS[8][128];
  const int w = threadIdx.x >> 5, lane = threadIdx.x & 31;
  const int node = blockIdx.x * 8 + w;  // grid sized exactly: N_NODES % 8 == 0

  if (lane < 25) obsS[w][lane] = obs[(size_t)node * 25 + lane];
  __syncthreads();

  // h = relu(LN(obs @ obs1 + b)); 4 hidden units per lane
  const int j0 = lane * 4;
  float a0 = obs1b[j0], a1 = obs1b[j0 + 1], a2 = obs1b[j0 + 2], a3 = obs1b[j0 + 3];
  for (int k = 0; k < 25; ++k) {
    const float o = obsS[w][k];
    const float4 wr = ((const float4*)(obs1w + k * 128))[lane];
    a0 += o * wr.x; a1 += o * wr.y; a2 += o * wr.z; a3 += o * wr.w;
  }
  const float mu = wave_sum32(a0 + a1 + a2 + a3) * (1.f / 128.f);
  const float d0 = a0 - mu, d1 = a1 - mu, d2 = a2 - mu, d3 = a3 - mu;
  const float var = wave_sum32(d0 * d0 + d1 * d1 + d2 * d2 + d3 * d3) * (1.f / 128.f);
  const float rs = rsqrtf(var + LN_EPS);
  float h0 = d0 * rs * obs_ln_g[j0]     + obs_ln_b[j0];
  float h1 = d1 * rs * obs_ln_g[j0 + 1] + obs_ln_b[j0 + 1];
  float h2 = d2 * rs * obs_ln_g[j0 + 2] + obs_ln_b[j0 + 2];
  float h3 = d3 * rs * obs_ln_g[j0 + 3] + obs_ln_b[j0 + 3];
  hS[w][j0]     = h0 > 0.f ? h0 : 0.f;
  hS[w][j0 + 1] = h1 > 0.f ? h1 : 0.f;
  hS[w][j0 + 2] = h2 > 0.f ? h2 : 0.f;
  hS[w][j0 + 3] = h3 > 0.f ? h3 : 0.f;

  // uncertainty branch (32 hidden, 1 per lane)
  float u = unc1b[lane];
  for (int k = 0; k < 5; ++k)
    u += unc[(size_t)node * 5 + k] * unc1w[k * 32 + lane];
  u1S[w][lane] = u > 0.f ? u : 0.f;
  __syncthreads();

  // obs_embed (2 outputs per lane)
  float oe0 = obs2b[lane], oe1 = obs2b[lane + 32];
  for (int j = 0; j < 128; ++j) {
    const float h = hS[w][j];
    oe0 += h * obs2w[j * 64 + lane];
    oe1 += h * obs2w[j * 64 + lane + 32];
  }
  float uw0 = unc2b[lane], uw1 = unc2b[lane + 32];
  for (int j = 0; j < 32; ++j) {
    const float uu = u1S[w][j];
    uw0 += uu * unc2w[j * 64 + lane];
    uw1 += uu * unc2w[j * 64 + lane + 32];
  }
  uw0 = 1.f / (1.f + expf(-uw0));
  uw1 = 1.f / (1.f + expf(-uw1));
  combS[w][lane]      = oe0;
  combS[w][lane + 32] = oe1;
  combS[w][64 + lane] = oe0 * (1.f - uw0);
  combS[w][96 + lane] = oe1 * (1.f - uw1);
  __syncthreads();

  // fusion + LN + relu
  float f0 = fusb[lane], f1 = fusb[lane + 32];
  for (int k = 0; k < 128; ++k) {
    const float c = combS[w][k];
    f0 += c * fusw[k * 64 + lane];
    f1 += c * fusw[k * 64 + lane + 32];
  }
  const float mu2 = wave_sum32(f0 + f1) * (1.f / 64.f);
  const float e0 = f0 - mu2, e1 = f1 - mu2;
  const float var2 = wave_sum32(e0 * e0 + e1 * e1) * (1.f / 64.f);
  const float rs2 = rsqrtf(var2 + LN_EPS);
  float y0 = e0 * rs2 * fus_ln_g[lane]      + fus_ln_b[lane];
  float y1 = e1 * rs2 * fus_ln_g[lane + 32] + fus_ln_b[lane + 32];
  xout[(size_t)node * 64 + lane]      = y0 > 0.f ? y0 : 0.f;
  xout[(size_t)node * 64 + lane + 32] = y1 > 0.f ? y1 : 0.f;
}

// ---------------------------------------------------------------------------
// Edge stage: one wave per 16-edge tile, 2 waves per block.
// ---------------------------------------------------------------------------
__global__ __launch_bounds__(64) void gnn_edge_kernel(
    const float* __restrict__ x,
    const int* __restrict__ src, const int* __restrict__ dst,
    const float* __restrict__ eattr,
    const float* __restrict__ pk_rel1, const float* __restrict__ rel1b,
    const float* __restrict__ rel2w, const float* __restrict__ rel2b,
    const float* __restrict__ pk_msg1, const float* __restrict__ msg1b,
    const float* __restrict__ pk_msg2, const float* __restrict__ msg2b,
    float* __restrict__ agg, float* __restrict__ counts, int n_tiles) {
  __shared__ __align__(16) float As[2][16 * LDA];
  __shared__ __align__(16) float Hs[2][16 * LDH];
  __shared__ float relS[2][16];
  __shared__ int   dstS[2][16];

  const int w    = threadIdx.x >> 5;
  const int lane = threadIdx.x & 31;
  const int tile = blockIdx.x * 2 + w;
  const bool active = tile < n_tiles;
  const int e0 = tile * 16;
  float* A = As[w];
  float* H = Hs[w];

  if (active) {
    if (lane < 16) dstS[w][lane] = dst[e0 + lane];
    // gather msg_in = [x_dst | x_src | edge_attr]; 2 lanes per edge, float4
    const int e = lane >> 1, p = lane & 1;
    const int ge = e0 + e;
    const int si = src[ge];
    const int di = dst[ge];
    float4* Arow = (float4*)(A + e * LDA);
    const float4* xd = (const float4*)(x + (size_t)di * EMB);
    const float4* xs = (const float4*)(x + (size_t)si * EMB);
    #pragma unroll
    for (int i = 0; i < 8; ++i) Arow[p * 8 + i] = xd[p * 8 + i];       // cols 0..63
    #pragma unroll
    for (int i = 0; i < 8; ++i) Arow[16 + p * 8 + i] = xs[p * 8 + i];  // cols 64..127
    Arow[32 + p] = ((const float4*)(eattr + (size_t)ge * EDGE_DIM))[p]; // cols 128..135
  }
  __syncthreads();

  // reliability: relu(A @ rel1_packed(swapped) + b) -> H
  if (active) wave_gemm_relu(A, LDA, (const v2f*)pk_rel1, rel1b, CATD, HID, H, LDH, lane);
  __syncthreads();
  if (active && lane < 16) {
    const float* Hrow = H + lane * LDH;
    float s = rel2b[0];
    for (int j = 0; j < HID; ++j) s += Hrow[j] * rel2w[j];
    relS[w][lane] = 1.f / (1.f + expf(-s));
  }
  __syncthreads();
  if (active && lane < 16) atomicAdd(&counts[dstS[w][lane]], relS[w][lane]);

  // messages: relu(A @ msg1 + b) -> H
  if (active) wave_gemm_relu(A, LDA, (const v2f*)pk_msg1, msg1b, CATD, HID, H, LDH, lane);
  __syncthreads();

  // msg2 + weight by attention(=0.25) * rel, scatter-add into agg[dst]
  if (active) {
    const int half = lane >> 4, nl = lane & 15;
    for (int nt = 0; nt < (EMB >> 4); ++nt) {
      v8f acc = wave_gemm_acc(H, LDH, (const v2f*)pk_msg2, HID >> 2, nt, lane);
      const float bs = msg2b[nt * 16 + nl];
      #pragma unroll
      for (int r = 0; r < 8; ++r) {
        const int m = r + 8 * half;
        const float val = (acc[r] + bs) * 0.25f * relS[w][m];
        atomicAdd(&agg[(size_t)dstS[w][m] * EMB + nt * 16 + nl], val);
      }
    }
  }
}

// ---------------------------------------------------------------------------
// Node stage: one wave per 16-node tile, 2 waves per block.
// ---------------------------------------------------------------------------
__global__ __launch_bounds__(64) void gnn_node_kernel(
    const float* __restrict__ x,
    const float* __restrict__ agg, const float* __restrict__ counts,
    const float* __restrict__ pk_self1, const float* __restrict__ self1b,
    const float* __restrict__ pk_self2, const float* __restrict__ self2b,
    const float* __restrict__ pk_upd1, const float* __restrict__ upd1b,
    const float* __restrict__ pk_upd2, const float* __restrict__ upd2b,
    const float* __restrict__ ln_g, const float* __restrict__ ln_b,
    float* __restrict__ xout, int n_tiles) {
  __shared__ __align__(16) float AUs[2][16 * LDN];
  __shared__ __align__(16) float Hs2[2][16 * LDN];
  __shared__ float Ys[2][16 * LDY];
  __shared__ float isoS[2][16];

  const int w    = threadIdx.x >> 5;
  const int lane = threadIdx.x & 31;
  const int tile = blockIdx.x * 2 + w;
  const bool active = tile < n_tiles;
  float* AU = AUs[w];
  float* H  = Hs2[w];
  float* Y  = Ys[w];

  if (active) {
    const int m = lane >> 1, p = lane & 1;
    const int node = tile * 16 + m;
    float4* Arow = (float4*)(AU + m * LDN);
    const float4* xr = (const float4*)(x + (size_t)node * EMB);
    const float4* ar = (const float4*)(agg + (size_t)node * EMB);
    #pragma unroll
    for (int i = 0; i < 8; ++i) Arow[p * 8 + i] = xr[p * 8 + i];       // cols 0..63 = x
    #pragma unroll
    for (int i = 0; i < 8; ++i) Arow[16 + p * 8 + i] = ar[p * 8 + i];  // cols 64..127 = agg
    if (lane < 16) isoS[w][lane] = expf(-counts[tile * 16 + lane]);
  }
  __syncthreads();

  // update path: H = relu([x|agg] @ upd1 + b)
  if (active) wave_gemm_relu(AU, LDN, (const v2f*)pk_upd1, upd1b, 2 * EMB, HID, H, LDN, lane);
  __syncthreads();
  if (active) {
    const int half = lane >> 4, nl = lane & 15;
    for (int nt = 0; nt < (EMB >> 4); ++nt) {
      v8f acc = wave_gemm_acc(H, LDN, (const v2f*)pk_upd2, HID >> 2, nt, lane);
      const float bs = upd2b[nt * 16 + nl];
      #pragma unroll
      for (int r = 0; r < 8; ++r)
        Y[(r + 8 * half) * LDY + nt * 16 + nl] = acc[r] + bs;  // raw update
    }
  }
  __syncthreads();

  // self path: H = relu(x @ self1 + b)   (x tile = AU cols 0..63, still intact)
  if (active) wave_gemm_relu(AU, LDN, (const v2f*)pk_self1, self1b, EMB, HID, H, LDN, lane);
  __syncthreads();
  if (active) {
    const int half = lane >> 4, nl = lane & 15;
    for (int nt = 0; nt < (EMB >> 4); ++nt) {
      v8f acc = wave_gemm_acc(H, LDN, (const v2f*)pk_self2, HID >> 2, nt, lane);
      const float bs = self2b[nt * 16 + nl];
      #pragma unroll
      for (int r = 0; r < 8; ++r) {
        const int m = r + 8 * half;
        const int n = nt * 16 + nl;
        const float self_v = acc[r] + bs;
        const float upd_v  = Y[m * LDY + n];
        const float iso    = isoS[w][m];
        Y[m * LDY + n] = AU[m * LDN + n] + (1.f - iso) * upd_v + iso * self_v;
      }
    }
  }
  __syncthreads();

  // LayerNorm per node (lanes 0..15 each own a node)
  if (active && lane < 16) {
    const float* yr = Y + lane * LDY;
    float mu = 0.f;
    for (int n = 0; n < EMB; ++n) mu += yr[n];
    mu *= (1.f / EMB);
    float var = 0.f;
    for (int n = 0; n < EMB; ++n) { const float d = yr[n] - mu; var += d * d; }
    var *= (1.f / EMB);
    const float rs = rsqrtf(var + LN_EPS);
    const size_t node = (size_t)tile * 16 + lane;
    for (int n = 0; n < EMB; ++n)
      xout[node * EMB + n] = (yr[n] - mu) * rs * ln_g[n] + ln_b[n];
  }
}

// ---------------------------------------------------------------------------
extern "C" void kernel_launch(void* const* d_in, const int* in_sizes, int n_in,
                              void* d_out, int out_size, void* d_ws, size_t ws_size,
                              hipStream_t stream) {
  (void)in_sizes; (void)n_in; (void)out_size; (void)ws_size;
  // top level in setup_inputs insertion order; params in jax tree order
  const float* obs   = (const float*)d_in[0];
  const float* unc   = (const float*)d_in[1];
  const int*   eidx  = (const int*)d_in[2];
  const float* eattr = (const float*)d_in[3];
  // enc: sorted keys {fus, fus_ln_b, fus_ln_g, obs1, obs2, obs_ln_b, obs_ln_g,
  //                   unc1, unc2}; each lin dict -> b then w
  const float* fus_b    = (const float*)d_in[4];
  const float* fus_w    = (const float*)d_in[5];
  const float* fus_ln_b = (const float*)d_in[6];
  const float* fus_ln_g = (const float*)d_in[7];
  const float* obs1_b   = (const float*)d_in[8];
  const float* obs1_w   = (const float*)d_in[9];
  const float* obs2_b   = (const float*)d_in[10];
  const float* obs2_w   = (const float*)d_in[11];
  const float* obs_ln_b = (const float*)d_in[12];
  const float* obs_ln_g = (const float*)d_in[13];
  const float* unc1_b   = (const float*)d_in[14];
  const float* unc1_w   = (const float*)d_in[15];
  const float* unc2_b   = (const float*)d_in[16];
  const float* unc2_w   = (const float*)d_in[17];

  struct Layer {
    const float *att1b, *att1w, *att2b, *att2w, *ln_b, *ln_g;
    const float *msg1b, *msg1w, *msg2b, *msg2w, *rel1b, *rel1w, *rel2b, *rel2w;
    const float *self1b, *self1w, *self2b, *self2w, *upd1b, *upd1w, *upd2b, *upd2w;
  } L[3];
  for (int i = 0; i < 3; ++i) {
    const int b = 18 + 22 * i;
    L[i].att1b  = (const float*)d_in[b + 0];  L[i].att1w  = (const float*)d_in[b + 1];
    L[i].att2b  = (const float*)d_in[b + 2];  L[i].att2w  = (const float*)d_in[b + 3];
    L[i].ln_b   = (const float*)d_in[b + 4];  L[i].ln_g   = (const float*)d_in[b + 5];
    L[i].msg1b  = (const float*)d_in[b + 6];  L[i].msg1w  = (const float*)d_in[b + 7];
    L[i].msg2b  = (const float*)d_in[b + 8];  L[i].msg2w  = (const float*)d_in[b + 9];
    L[i].rel1b  = (const float*)d_in[b + 10]; L[i].rel1w  = (const float*)d_in[b + 11];
    L[i].rel2b  = (const float*)d_in[b + 12]; L[i].rel2w  = (const float*)d_in[b + 13];
    L[i].self1b = (const float*)d_in[b + 14]; L[i].self1w = (const float*)d_in[b + 15];
    L[i].self2b = (const float*)d_in[b + 16]; L[i].self2w = (const float*)d_in[b + 17];
    L[i].upd1b  = (const float*)d_in[b + 18]; L[i].upd1w  = (const float*)d_in[b + 19];
    L[i].upd2b  = (const float*)d_in[b + 20]; L[i].upd2w  = (const float*)d_in[b + 21];
  }

  // workspace (floats): x0 | x1 | agg | counts | packed weights x3 layers (~40 MB)
  const size_t SZ_REL1 = (size_t)CATD * HID, SZ_MSG1 = (size_t)CATD * HID;
  const size_t SZ_MSG2 = (size_t)HID * EMB;
  const size_t SZ_UPD1 = (size_t)(2 * EMB) * HID, SZ_UPD2 = (size_t)HID * EMB;
  const size_t SZ_SELF1 = (size_t)EMB * HID, SZ_SELF2 = (size_t)HID * EMB;
  const size_t PL = SZ_REL1 + SZ_MSG1 + SZ_MSG2 + SZ_UPD1 + SZ_UPD2 + SZ_SELF1 + SZ_SELF2;

  float* ws  = (float*)d_ws;
  float* x0  = ws;
  float* x1  = x0 + (size_t)N_NODES * EMB;
  float* agg = x1 + (size_t)N_NODES * EMB;
  float* cnt = agg + (size_t)N_NODES * EMB;
  float* pkw = cnt + N_NODES;

  float *pk_rel1[3], *pk_msg1[3], *pk_msg2[3], *pk_upd1[3], *pk_upd2[3],
        *pk_self1[3], *pk_self2[3];
  for (int i = 0; i < 3; ++i) {
    float* base = pkw + (size_t)i * PL;
    pk_rel1[i]  = base;                       base += SZ_REL1;
    pk_msg1[i]  = base;                       base += SZ_MSG1;
    pk_msg2[i]  = base;                       base += SZ_MSG2;
    pk_upd1[i]  = base;                       base += SZ_UPD1;
    pk_upd2[i]  = base;                       base += SZ_UPD2;
    pk_self1[i] = base;                       base += SZ_SELF1;
    pk_self2[i] = base;
  }

  // one-shot weight packing into WMMA fragment order
  #define PACK(src_, dst_, kd_, no_, swap_)                                     \
    gnn_pack_weights_kernel<<<((kd_) * (no_) + 255) / 256, 256, 0, stream>>>(   \
        (src_), (dst_), (kd_), (no_), (swap_))
  for (int i = 0; i < 3; ++i) {
    PACK(L[i].rel1w,  pk_rel1[i],  CATD,    HID, 1);  // fold src/dst half swap
    PACK(L[i].msg1w,  pk_msg1[i],  CATD,    HID, 0);
    PACK(L[i].msg2w,  pk_msg2[i],  HID,     EMB, 0);
    PACK(L[i].upd1w,  pk_upd1[i],  2 * EMB, HID, 0);
    PACK(L[i].upd2w,  pk_upd2[i],  HID,     EMB, 0);
    PACK(L[i].self1w, pk_self1[i], EMB,     HID, 0);
    PACK(L[i].self2w, pk_self2[i], HID,     EMB, 0);
  }
  #undef PACK

  gnn_encoder_kernel<<<N_NODES / 8, 256, 0, stream>>>(
      obs, unc, obs1_w, obs1_b, obs_ln_g, obs_ln_b, obs2_w, obs2_b,
      unc1_w, unc1_b, unc2_w, unc2_b, fus_w, fus_b, fus_ln_g, fus_ln_b, x0);

  const int* srcp = eidx;
  const int* dstp = eidx + N_EDGES;
  const int etiles = N_EDGES / 16;  // 50000
  const int ntiles = N_NODES / 16;  // 3125
  float* xin = x0;
  for (int i = 0; i < 3; ++i) {
    gnn_zero_kernel<<<2048, 256, 0, stream>>>(agg, (size_t)N_NODES * EMB + N_NODES);
    gnn_edge_kernel<<<(etiles + 1) / 2, 64, 0, stream>>>(
        xin, srcp, dstp, eattr,
        pk_rel1[i], L[i].rel1b, L[i].rel2w, L[i].rel2b,
        pk_msg1[i], L[i].msg1b, pk_msg2[i], L[i].msg2b, agg, cnt, etiles);
    float* xo = (i == 2) ? (float*)d_out : ((xin == x0) ? x1 : x0);
    gnn_node_kernel<<<(ntiles + 1) / 2, 64, 0, stream>>>(
        xin, agg, cnt,
        pk_self1[i], L[i].self1b, pk_self2[i], L[i].self2b,
        pk_upd1[i], L[i].upd1b, pk_upd2[i], L[i].upd2b,
        L[i].ln_g, L[i].ln_b, xo, ntiles);
    xin = xo;
  }
}